// MolecularGNN_48249662603743
// MI455X (gfx1250) — compile-verified
//
#include <hip/hip_runtime.h>
#include <hip/hip_bf16.h>

#define N_NODES 50000
#define N_EDGES 800000
#define IN_CH   128
#define HID     256
#define N_GRAPHS 1024
#define BN_EPS  1e-5f

typedef float v2f __attribute__((ext_vector_type(2)));
typedef float v8f __attribute__((ext_vector_type(8)));

// ---------------------------------------------------------------------------
// Generic fill
__global__ void k_fill(float* p, float v, size_t n) {
    size_t i = (size_t)blockIdx.x * blockDim.x + threadIdx.x;
    if (i < n) p[i] = v;
}

// deg[dst[e]] += 1
__global__ void k_deg_accum(const int* dst, float* deg, int E) {
    int e = blockIdx.x * blockDim.x + threadIdx.x;
    if (e < E) atomicAdd(&deg[dst[e]], 1.0f);
}

// deg -> rsqrt(deg) in place
__global__ void k_dinv(float* deg, int n) {
    int i = blockIdx.x * blockDim.x + threadIdx.x;
    if (i < n) { float d = deg[i]; deg[i] = d > 0.0f ? rsqrtf(d) : 0.0f; }
}

// ---------------------------------------------------------------------------
// WMMA fp32 GEMM: C[M,N] = A[M,K] @ B[K,N] (+ bias[n] if bias != null)
// One wave per 16x16 output tile. Requires N % 16 == 0, K % 4 == 0.
__global__ void k_gemm_wmma_f32(const float* __restrict__ A,
                                const float* __restrict__ B,
                                const float* __restrict__ bias,
                                float* __restrict__ C,
                                int M, int N, int K) {
    int wave = (int)((blockIdx.x * blockDim.x + threadIdx.x) >> 5);
    int lane = threadIdx.x & 31;
    int ntiles = N >> 4;
    int mtiles = (M + 15) >> 4;
    if (wave >= ntiles * mtiles) return;            // wave-uniform exit (EXEC stays full)
    int tileM = wave / ntiles;
    int tileN = wave - tileM * ntiles;

    int m = tileM * 16 + (lane & 15);               // A row for this lane
    int n = tileN * 16 + (lane & 15);               // B/C col for this lane
    int khalf = (lane >> 4) * 2;                    // lanes 16-31 handle K+2,K+3

    v8f c = {};
    const float* arow = A + (size_t)m * K;
    for (int k = 0; k < K; k += 4) {
        v2f a, b;
        const float* ap = arow + k + khalf;
        a.x = ap[0];
        a.y = ap[1];
        const float* bp = B + (size_t)(k + khalf) * N + n;
        b.x = bp[0];
        b.y = bp[N];
        c = __builtin_amdgcn_wmma_f32_16x16x4_f32(false, a, false, b,
                                                  (short)0, c, false, false);
    }

    float bn = bias ? bias[n] : 0.0f;
    int rbase = tileM * 16 + 8 * (lane >> 4);       // VGPR r -> row rbase+r
#pragma unroll
    for (int r = 0; r < 8; ++r) {
        int row = rbase + r;
        if (row < M) C[(size_t)row * N + n] = c[r] + bn;
    }
}

// ---------------------------------------------------------------------------
// out[n][c] = bias[c]   (row-broadcast init before scatter accumulation)
__global__ void k_init_rows_bias(float* out, const float* bias, size_t total) {
    size_t i = (size_t)blockIdx.x * blockDim.x + threadIdx.x;
    if (i < total) out[i] = bias[i & (HID - 1)];
}

// One wave per edge: out[dst] += hw[src] * dinv[src]*dinv[dst]  (256 channels)
__global__ void k_edge_scatter(const float* __restrict__ hw,
                               const int* __restrict__ src,
                               const int* __restrict__ dst,
                               const float* __restrict__ dinv,
                               float* __restrict__ out, int E) {
    int e = (int)((blockIdx.x * blockDim.x + threadIdx.x) >> 5);
    int lane = threadIdx.x & 31;
    if (e >= E) return;
    int s = src[e], d = dst[e];
    float norm = dinv[s] * dinv[d];
    const float4* hp = (const float4*)(hw + (size_t)s * HID);
    float* op = out + (size_t)d * HID;
#pragma unroll
    for (int i = 0; i < 2; ++i) {
        float4 v = hp[lane + 32 * i];
        int base = (lane + 32 * i) * 4;
        atomicAdd(op + base + 0, v.x * norm);
        atomicAdd(op + base + 1, v.y * norm);
        atomicAdd(op + base + 2, v.z * norm);
        atomicAdd(op + base + 3, v.w * norm);
    }
}

// Self-loop: out[n][:] += hw[n][:] * dinv[n]^2   (single writer per row -> no atomics)
__global__ void k_self_loop(const float* __restrict__ hw,
                            const float* __restrict__ dinv,
                            float* __restrict__ out, size_t total) {
    size_t i = (size_t)blockIdx.x * blockDim.x + threadIdx.x;
    if (i < total) {
        int n = (int)(i >> 8);
        float di = dinv[n];
        out[i] += hw[i] * di * di;
    }
}

// ---------------------------------------------------------------------------
// BN stats: per-channel partial sum / sumsq, coalesced rows, atomic combine
__global__ void k_bn_stats(const float* __restrict__ h, float* sum, float* sumsq,
                           int rows) {
    int c = threadIdx.x;                 // blockDim.x == HID
    float s = 0.0f, s2 = 0.0f;
    for (int n = blockIdx.x; n < rows; n += gridDim.x) {
        float v = h[(size_t)n * HID + c];
        s += v; s2 += v * v;
    }
    atomicAdd(&sum[c], s);
    atomicAdd(&sumsq[c], s2);
}

__global__ void k_bn_finalize(const float* sum, const float* sumsq,
                              const float* gamma, const float* beta,
                              float* sscale, float* sshift, float inv_n) {
    int c = threadIdx.x;
    float mu = sum[c] * inv_n;
    float var = sumsq[c] * inv_n - mu * mu;
    float istd = rsqrtf(var + BN_EPS);
    float sc = gamma[c] * istd;
    sscale[c] = sc;
    sshift[c] = beta[c] - mu * sc;
}

__global__ void k_bn_apply_relu(float* h, const float* sscale, const float* sshift,
                                size_t total) {
    size_t i = (size_t)blockIdx.x * blockDim.x + threadIdx.x;
    if (i < total) {
        int c = (int)(i & (HID - 1));
        float v = h[i] * sscale[c] + sshift[c];
        h[i] = v > 0.0f ? v : 0.0f;
    }
}

// ---------------------------------------------------------------------------
// Mean pool: one wave per node, atomic row add into pooled[batch[n]]
__global__ void k_pool(const float* __restrict__ h, const int* __restrict__ batch,
                       float* pooled, float* counts, int rows) {
    int nidx = (int)((blockIdx.x * blockDim.x + threadIdx.x) >> 5);
    int lane = threadIdx.x & 31;
    if (nidx >= rows) return;
    int g = batch[nidx];
    const float* hp = h + (size_t)nidx * HID;
    float* pp = pooled + (size_t)g * HID;
#pragma unroll
    for (int i = 0; i < 8; ++i) {
        int c = lane + 32 * i;
        atomicAdd(pp + c, hp[c]);
    }
    if (lane == 0) atomicAdd(counts + g, 1.0f);
}

__global__ void k_pool_div(float* pooled, const float* counts, size_t total) {
    size_t i = (size_t)blockIdx.x * blockDim.x + threadIdx.x;
    if (i < total) {
        float cnt = counts[i >> 8];
        pooled[i] *= 1.0f / fmaxf(cnt, 1.0f);
    }
}

// ---------------------------------------------------------------------------
// Final matvec: out[g] = dot(z[g,:], W2[:,0]) + b2   (one wave per graph)
__global__ void k_final_mv(const float* __restrict__ z, const float* __restrict__ W2,
                           const float* __restrict__ b2, float* out, int G) {
    int g = (int)((blockIdx.x * blockDim.x + threadIdx.x) >> 5);
    int lane = threadIdx.x & 31;
    if (g >= G) return;
    float acc = 0.0f;
    const float* zr = z + (size_t)g * HID;
#pragma unroll
    for (int i = 0; i < 8; ++i) acc += zr[lane + 32 * i] * W2[lane + 32 * i];
#pragma unroll
    for (int off = 16; off > 0; off >>= 1) acc += __shfl_down(acc, off, 32);
    if (lane == 0) out[g] = acc + b2[0];
}

// ---------------------------------------------------------------------------
extern "C" void kernel_launch(void* const* d_in, const int* in_sizes, int n_in,
                              void* d_out, int out_size, void* d_ws, size_t ws_size,
                              hipStream_t stream) {
    const float* x      = (const float*)d_in[0];
    const int*   ei     = (const int*)  d_in[1];   // [2, E] (int32 under jax defaults)
    const int*   batch  = (const int*)  d_in[2];
    const float* W_emb  = (const float*)d_in[3];
    const float* b_emb  = (const float*)d_in[4];
    const float* W_conv = (const float*)d_in[5];   // [3, HID, HID]
    const float* b_conv = (const float*)d_in[6];   // [3, HID]
    const float* bn_g   = (const float*)d_in[7];
    const float* bn_b   = (const float*)d_in[8];
    const float* W1     = (const float*)d_in[9];
    const float* b1     = (const float*)d_in[10];
    const float* g2     = (const float*)d_in[11];
    const float* bt2    = (const float*)d_in[12];
    const float* W2     = (const float*)d_in[13];
    const float* b2     = (const float*)d_in[14];
    float* out = (float*)d_out;

    const int* src = ei;                // edge_index[0]
    const int* dst = ei + N_EDGES;      // edge_index[1]

    // ---- workspace carving ----
    size_t off = 0;
    auto alloc = [&](size_t bytes) {
        void* p = (char*)d_ws + off;
        off += (bytes + 255) & ~(size_t)255;
        return p;
    };
    float* H      = (float*)alloc((size_t)N_NODES * HID * 4);
    float* T      = (float*)alloc((size_t)N_NODES * HID * 4);
    float* dinv   = (float*)alloc((size_t)N_NODES * 4);
    float* ssum   = (float*)alloc(HID * 4);
    float* ssq    = (float*)alloc(HID * 4);
    float* sscale = (float*)alloc(HID * 4);
    float* sshift = (float*)alloc(HID * 4);
    float* pooled = (float*)alloc((size_t)N_GRAPHS * HID * 4);
    float* counts = (float*)alloc((size_t)N_GRAPHS * 4);
    float* Z      = (float*)alloc((size_t)N_GRAPHS * HID * 4);
    (void)ws_size;

    const size_t NH = (size_t)N_NODES * HID;   // 12.8M
    const size_t GH = (size_t)N_GRAPHS * HID;  // 256K
    auto blk = [](size_t n, int t) { return (unsigned)((n + t - 1) / t); };

    // ---- degree / dinv (self-loop => init 1) ----
    k_fill<<<blk(N_NODES, 256), 256, 0, stream>>>(dinv, 1.0f, N_NODES);
    k_deg_accum<<<blk(N_EDGES, 256), 256, 0, stream>>>(dst, dinv, N_EDGES);
    k_dinv<<<blk(N_NODES, 256), 256, 0, stream>>>(dinv, N_NODES);

    // ---- embedding GEMM: H = x @ W_emb + b_emb ----
    {
        size_t waves = (size_t)(N_NODES / 16) * (HID / 16);
        k_gemm_wmma_f32<<<blk(waves * 32, 256), 256, 0, stream>>>(
            x, W_emb, b_emb, H, N_NODES, HID, IN_CH);
    }

    // ---- 3 GCN layers ----
    for (int l = 0; l < 3; ++l) {
        size_t waves = (size_t)(N_NODES / 16) * (HID / 16);
        // T = H @ W_conv[l]
        k_gemm_wmma_f32<<<blk(waves * 32, 256), 256, 0, stream>>>(
            H, W_conv + (size_t)l * HID * HID, nullptr, T, N_NODES, HID, HID);
        // H := b_conv[l] broadcast, then scatter-add messages
        k_init_rows_bias<<<blk(NH, 256), 256, 0, stream>>>(H, b_conv + l * HID, NH);
        k_edge_scatter<<<blk((size_t)N_EDGES * 32, 256), 256, 0, stream>>>(
            T, src, dst, dinv, H, N_EDGES);
        k_self_loop<<<blk(NH, 256), 256, 0, stream>>>(T, dinv, H, NH);
        // BN + ReLU
        k_fill<<<1, 256, 0, stream>>>(ssum, 0.0f, HID);
        k_fill<<<1, 256, 0, stream>>>(ssq, 0.0f, HID);
        k_bn_stats<<<512, HID, 0, stream>>>(H, ssum, ssq, N_NODES);
        k_bn_finalize<<<1, HID, 0, stream>>>(ssum, ssq, bn_g + l * HID, bn_b + l * HID,
                                             sscale, sshift, 1.0f / N_NODES);
        k_bn_apply_relu<<<blk(NH, 256), 256, 0, stream>>>(H, sscale, sshift, NH);
    }

    // ---- global mean pool ----
    k_fill<<<blk(GH, 256), 256, 0, stream>>>(pooled, 0.0f, GH);
    k_fill<<<blk(N_GRAPHS, 256), 256, 0, stream>>>(counts, 0.0f, N_GRAPHS);
    k_pool<<<blk((size_t)N_NODES * 32, 256), 256, 0, stream>>>(H, batch, pooled, counts, N_NODES);
    k_pool_div<<<blk(GH, 256), 256, 0, stream>>>(pooled, counts, GH);

    // ---- head: Z = pooled @ W1 + b1; BN; ReLU; out = Z @ W2 + b2 ----
    {
        size_t waves = (size_t)(N_GRAPHS / 16) * (HID / 16);
        k_gemm_wmma_f32<<<blk(waves * 32, 256), 256, 0, stream>>>(
            pooled, W1, b1, Z, N_GRAPHS, HID, HID);
    }
    k_fill<<<1, 256, 0, stream>>>(ssum, 0.0f, HID);
    k_fill<<<1, 256, 0, stream>>>(ssq, 0.0f, HID);
    k_bn_stats<<<128, HID, 0, stream>>>(Z, ssum, ssq, N_GRAPHS);
    k_bn_finalize<<<1, HID, 0, stream>>>(ssum, ssq, g2, bt2, sscale, sshift,
                                         1.0f / N_GRAPHS);
    k_bn_apply_relu<<<blk(GH, 256), 256, 0, stream>>>(Z, sscale, sshift, GH);
    k_final_mv<<<blk((size_t)N_GRAPHS * 32, 256), 256, 0, stream>>>(Z, W2, b2, out, N_GRAPHS);
}